// AdditiveAttention_10402410791014
// MI455X (gfx1250) — compile-verified
//
#include <hip/hip_runtime.h>
#include <math.h>

typedef __attribute__((ext_vector_type(2))) float v2f;
typedef __attribute__((ext_vector_type(8))) float v8f;

#define B_   8
#define Q_   512
#define K_   1024
#define D_   128     // DQ = DK = DV
#define H_   64
#define QT_  8       // q-rows per score block
#define MASK_VAL_ (-1000000.0f)

#if __has_builtin(__builtin_amdgcn_wmma_f32_16x16x4_f32)
#define HAVE_WMMA_F32 1
#endif

static __device__ __forceinline__ float fast_tanh(float x) {
#if __has_builtin(__builtin_amdgcn_tanhf)
  return __builtin_amdgcn_tanhf(x);
#else
  float r;
  asm volatile("v_tanh_f32 %0, %1" : "=v"(r) : "v"(x));
  return r;
#endif
}

// ---------------------------------------------------------------------------
// Y[rows, 64] = X[rows, 128] @ W[128, 64]   (rows = B*Q or B*K, flattened)
// 128 threads = 4 waves; block = one 16-row M tile, wave w = N-tile w.
// v_wmma_f32_16x16x4_f32, K stepped by 4.
// ---------------------------------------------------------------------------
__global__ __launch_bounds__(128) void proj_kernel(const float* __restrict__ X,
                                                   const float* __restrict__ W,
                                                   float* __restrict__ Y) {
  const int m0   = blockIdx.x * 16;
  const int wave = threadIdx.x >> 5;
  const int lane = threadIdx.x & 31;
  const int n0   = wave * 16;
  const int mn   = lane & 15;
  const int koff = (lane >> 4) * 2;
#if defined(HAVE_WMMA_F32)
  v8f c = {};
  const float* xrow = X + (size_t)(m0 + mn) * D_;
  for (int k0 = 0; k0 < D_; k0 += 4) {
    v2f a, b;
    a.x = xrow[k0 + koff];
    a.y = xrow[k0 + koff + 1];
    b.x = W[(size_t)(k0 + koff)     * H_ + n0 + mn];
    b.y = W[(size_t)(k0 + koff + 1) * H_ + n0 + mn];
    c = __builtin_amdgcn_wmma_f32_16x16x4_f32(false, a, false, b, (short)0, c, false, false);
  }
  const int mbase = (lane >> 4) * 8;   // VGPR r: lanes0-15 -> M=r, lanes16-31 -> M=8+r
  #pragma unroll
  for (int r = 0; r < 8; ++r)
    Y[(size_t)(m0 + mbase + r) * H_ + n0 + mn] = c[r];
#else
  for (int idx = threadIdx.x; idx < 16 * H_; idx += 128) {
    const int m = m0 + idx / H_;
    const int n = idx % H_;
    float s = 0.f;
    for (int k = 0; k < D_; ++k) s += X[(size_t)m * D_ + k] * W[(size_t)k * H_ + n];
    Y[(size_t)m * H_ + n] = s;
  }
#endif
}

// ---------------------------------------------------------------------------
// Fused score + masked softmax for an 8-row q tile of one batch.
// scores[q,k] = sum_h w_v[h] * tanh(qh[b,q,h] + kh[b,k,h]);  k >= vlen -> -1e6
// 256 threads: score phase — thread t owns k = t, t+256, ... and accumulates
// all 8 q rows per kh read (8x global-read reuse). Softmax phase — wave w
// owns row w, wave32 __shfl_xor reductions, no cross-wave barriers.
// ---------------------------------------------------------------------------
__global__ __launch_bounds__(256) void score_softmax_kernel(
    const float* __restrict__ qh,          // [B,Q,H]
    const float* __restrict__ kh,          // [B,K,H]
    const float* __restrict__ wv,          // [H]
    const int*   __restrict__ valid_lens,  // [B]
    float* __restrict__ attn)              // [B,Q,K]
{
  __shared__ float s_q[QT_ * H_];          // 2 KB
  __shared__ float s_wv[H_];               // 256 B
  __shared__ float s_sc[QT_][K_];          // 32 KB

  const int b    = blockIdx.y;
  const int q0   = blockIdx.x * QT_;
  const int tid  = threadIdx.x;
  const int vlen = valid_lens[b];

  // qh rows q0..q0+7 are contiguous in memory: flat copy
  for (int i = tid; i < QT_ * H_; i += 256)
    s_q[i] = qh[((size_t)b * Q_ + q0) * H_ + i];
  if (tid < H_) s_wv[tid] = wv[tid];
  __syncthreads();

  const float*  khb   = kh + (size_t)b * K_ * H_;
  const float4* s_q4  = (const float4*)s_q;
  const float4* s_wv4 = (const float4*)s_wv;

  #pragma unroll 1
  for (int kk = 0; kk < K_ / 256; ++kk) {
    const int k = kk * 256 + tid;
    float acc[QT_];
    if (k < vlen) {
      #pragma unroll
      for (int qi = 0; qi < QT_; ++qi) acc[qi] = 0.f;
      const float4* kr = (const float4*)(khb + (size_t)k * H_);
      #pragma unroll 4
      for (int h4 = 0; h4 < H_ / 4; ++h4) {
        const float4 kv = kr[h4];
        const float4 w4 = s_wv4[h4];
        #pragma unroll
        for (int qi = 0; qi < QT_; ++qi) {
          const float4 qv = s_q4[qi * (H_ / 4) + h4];
          acc[qi] += w4.x * fast_tanh(qv.x + kv.x);
          acc[qi] += w4.y * fast_tanh(qv.y + kv.y);
          acc[qi] += w4.z * fast_tanh(qv.z + kv.z);
          acc[qi] += w4.w * fast_tanh(qv.w + kv.w);
        }
      }
    } else {
      #pragma unroll
      for (int qi = 0; qi < QT_; ++qi) acc[qi] = MASK_VAL_;
    }
    #pragma unroll
    for (int qi = 0; qi < QT_; ++qi) s_sc[qi][k] = acc[qi];
  }
  __syncthreads();

  // per-wave masked softmax: wave w -> row w
  const int wrow = tid >> 5;
  const int lane = tid & 31;
  float* row = s_sc[wrow];

  float lmax = -INFINITY;
  #pragma unroll
  for (int i = 0; i < K_ / 32; ++i) lmax = fmaxf(lmax, row[lane + 32 * i]);
  #pragma unroll
  for (int off = 16; off > 0; off >>= 1)
    lmax = fmaxf(lmax, __shfl_xor(lmax, off, 32));

  float lsum = 0.f;
  #pragma unroll
  for (int i = 0; i < K_ / 32; ++i) {
    const float e = __expf(row[lane + 32 * i] - lmax);
    row[lane + 32 * i] = e;
    lsum += e;
  }
  #pragma unroll
  for (int off = 16; off > 0; off >>= 1)
    lsum += __shfl_xor(lsum, off, 32);
  const float inv = 1.0f / lsum;

  float* arow = attn + ((size_t)b * Q_ + q0 + wrow) * K_;
  #pragma unroll
  for (int i = 0; i < K_ / 32; ++i)
    arow[lane + 32 * i] = row[lane + 32 * i] * inv;
}

// ---------------------------------------------------------------------------
// out[b] = attn[b] @ values[b] : [512,1024]@[1024,128] f32 via WMMA 16x16x4.
// 256 threads = 8 waves; wave w owns N-tile w (DV = 128 = 8*16);
// block owns one 16-row M tile of one batch.
// ---------------------------------------------------------------------------
__global__ __launch_bounds__(256) void av_wmma_kernel(
    const float* __restrict__ attn,   // [B,Q,K]
    const float* __restrict__ V,      // [B,K,DV]
    float* __restrict__ out)          // [B,Q,DV]
{
  const int b    = blockIdx.y;
  const int m0   = blockIdx.x * 16;
  const int wave = threadIdx.x >> 5;
  const int lane = threadIdx.x & 31;
  const int n0   = wave * 16;
  const int mn   = lane & 15;
  const int koff = (lane >> 4) * 2;
#if defined(HAVE_WMMA_F32)
  v8f c = {};
  const float* arow = attn + ((size_t)b * Q_ + (m0 + mn)) * K_;
  const float* vb   = V + (size_t)b * K_ * D_;
  for (int k0 = 0; k0 < K_; k0 += 4) {
    v2f a, bb;
    a.x  = arow[k0 + koff];
    a.y  = arow[k0 + koff + 1];
    bb.x = vb[(size_t)(k0 + koff)     * D_ + n0 + mn];
    bb.y = vb[(size_t)(k0 + koff + 1) * D_ + n0 + mn];
    c = __builtin_amdgcn_wmma_f32_16x16x4_f32(false, a, false, bb, (short)0, c, false, false);
  }
  const int mbase = (lane >> 4) * 8;
  #pragma unroll
  for (int r = 0; r < 8; ++r)
    out[((size_t)b * Q_ + (m0 + mbase + r)) * D_ + n0 + mn] = c[r];
#else
  for (int idx = threadIdx.x; idx < 16 * D_; idx += 256) {
    const int m = m0 + idx / D_;
    const int n = idx % D_;
    float s = 0.f;
    const float* ar = attn + ((size_t)b * Q_ + m) * K_;
    for (int k = 0; k < K_; ++k) s += ar[k] * V[((size_t)b * K_ + k) * D_ + n];
    out[((size_t)b * Q_ + m) * D_ + n] = s;
  }
#endif
}

extern "C" void kernel_launch(void* const* d_in, const int* in_sizes, int n_in,
                              void* d_out, int out_size, void* d_ws, size_t ws_size,
                              hipStream_t stream) {
  const float* queries = (const float*)d_in[0];
  const float* keys    = (const float*)d_in[1];
  const float* values  = (const float*)d_in[2];
  const int*   vlens   = (const int*)d_in[3];
  const float* W_q     = (const float*)d_in[4];
  const float* W_k     = (const float*)d_in[5];
  const float* w_v     = (const float*)d_in[6];
  float* out = (float*)d_out;

  // workspace layout (floats): qh [B*Q*H] | kh [B*K*H] | attn [B*Q*K]
  float* qh   = (float*)d_ws;
  float* kh   = qh + (size_t)B_ * Q_ * H_;
  float* attn = kh + (size_t)B_ * K_ * H_;

  proj_kernel<<<(B_ * Q_) / 16, 128, 0, stream>>>(queries, W_q, qh);
  proj_kernel<<<(B_ * K_) / 16, 128, 0, stream>>>(keys,    W_k, kh);

  dim3 g1(Q_ / QT_, B_);
  score_softmax_kernel<<<g1, 256, 0, stream>>>(qh, kh, w_v, vlens, attn);

  dim3 g2(Q_ / 16, B_);
  av_wmma_kernel<<<g2, 256, 0, stream>>>(attn, values, out);
}